// SequenceTransformer_50002009260083
// MI455X (gfx1250) — compile-verified
//
#include <hip/hip_runtime.h>
#include <math.h>

typedef __attribute__((ext_vector_type(2))) float v2f;
typedef __attribute__((ext_vector_type(8))) float v8f;

#define SEQ     8
#define DFF     32
#define NLAYER  2
#define VOCAB   256
#define BLOCK   256

// Branch-free erf, Abramowitz & Stegun 7.1.26, |err| <= 1.5e-7.
// Uses v_rcp_f32 / v_exp_f32 (1-ulp HW transcendentals), no range branches.
__device__ __forceinline__ float erf_fast(float v) {
    const float ax = fabsf(v);
    const float t  = __builtin_amdgcn_rcpf(fmaf(0.3275911f, ax, 1.0f));
    float p = fmaf(1.061405429f, t, -1.453152027f);
    p = fmaf(p, t, 1.421413741f);
    p = fmaf(p, t, -0.284496736f);
    p = fmaf(p, t, 0.254829592f);
    p = p * t;
    // exp(-ax^2) = exp2(-ax^2 * log2(e))
    const float e = __builtin_amdgcn_exp2f(ax * ax * -1.4426950408889634f);
    const float r = fmaf(-p, e, 1.0f);
    return __builtin_copysignf(r, v);
}

__device__ __forceinline__ float gelu_exact(float t) {
    return 0.5f * t * (1.0f + erf_fast(t * 0.70710678118654752f));
}

__global__ __launch_bounds__(BLOCK) void txf_fused_kernel(
    const int*   __restrict__ x,
    const float* __restrict__ Wqkv, const float* __restrict__ bqkv,
    const float* __restrict__ Wo,   const float* __restrict__ bo,
    const float* __restrict__ W1,   const float* __restrict__ b1,
    const float* __restrict__ W2,   const float* __restrict__ b2,
    const float* __restrict__ g1,   const float* __restrict__ be1,
    const float* __restrict__ g2,   const float* __restrict__ be2,
    const float* __restrict__ Wout, const float* __restrict__ bout,
    float* __restrict__ out)
{
    __shared__ float sfinal[BLOCK * 2];

    const int tid = threadIdx.x;
    const int b   = blockIdx.x * BLOCK + tid;

    // ---------------- phase 1: tiny transformer, one batch row per thread ----
    float h0[SEQ], h1[SEQ];
    {
        // 32 contiguous bytes per thread -> two B128 loads
        const int4* xp = (const int4*)(x + b * SEQ);
        int4 xa = xp[0];
        int4 xb = xp[1];
        int xi[SEQ] = { xa.x, xa.y, xa.z, xa.w, xb.x, xb.y, xb.z, xb.w };
#pragma unroll
        for (int s = 0; s < SEQ; ++s) {
            h0[s] = (float)xi[s] * (1.0f / 255.0f);
            h1[s] = (float)s * (1.0f / 7.0f);
        }
    }

#pragma unroll
    for (int l = 0; l < NLAYER; ++l) {
        float q[SEQ][2], k[SEQ][2], v[SEQ][2];
#pragma unroll
        for (int s = 0; s < SEQ; ++s) {
            const float c0 = h0[s], c1 = h1[s];
            const int wb = l * 12, bb_ = l * 6;
            q[s][0] = Wqkv[wb + 0] * c0 + Wqkv[wb + 1] * c1 + bqkv[bb_ + 0];
            q[s][1] = Wqkv[wb + 2] * c0 + Wqkv[wb + 3] * c1 + bqkv[bb_ + 1];
            k[s][0] = Wqkv[wb + 4] * c0 + Wqkv[wb + 5] * c1 + bqkv[bb_ + 2];
            k[s][1] = Wqkv[wb + 6] * c0 + Wqkv[wb + 7] * c1 + bqkv[bb_ + 3];
            v[s][0] = Wqkv[wb + 8] * c0 + Wqkv[wb + 9] * c1 + bqkv[bb_ + 4];
            v[s][1] = Wqkv[wb +10] * c0 + Wqkv[wb +11] * c1 + bqkv[bb_ + 5];
        }

        // attention: H=2 heads, head_dim=1 -> scale = 1/sqrt(1) = 1
        float o[SEQ][2];
#pragma unroll
        for (int hh = 0; hh < 2; ++hh) {
#pragma unroll
            for (int qs = 0; qs < SEQ; ++qs) {
                float sc[SEQ];
                float m = -1e30f;
#pragma unroll
                for (int ks = 0; ks < SEQ; ++ks) {
                    sc[ks] = q[qs][hh] * k[ks][hh];
                    m = fmaxf(m, sc[ks]);
                }
                float den = 0.0f, num = 0.0f;
#pragma unroll
                for (int ks = 0; ks < SEQ; ++ks) {
                    float e = __expf(sc[ks] - m);
                    den += e;
                    num += e * v[ks][hh];
                }
                o[qs][hh] = num * __builtin_amdgcn_rcpf(den);
            }
        }

#pragma unroll
        for (int s = 0; s < SEQ; ++s) {
            // Wo projection + residual
            float y0 = Wo[l*4+0]*o[s][0] + Wo[l*4+1]*o[s][1] + bo[l*2+0] + h0[s];
            float y1 = Wo[l*4+2]*o[s][0] + Wo[l*4+3]*o[s][1] + bo[l*2+1] + h1[s];
            // layernorm over 2 elements (v_rsq_f32)
            float mu = 0.5f * (y0 + y1);
            float d0 = y0 - mu, d1 = y1 - mu;
            float inv = __builtin_amdgcn_rsqf(0.5f * (d0*d0 + d1*d1) + 1e-5f);
            float a0 = d0 * inv * g1[l*2+0] + be1[l*2+0];
            float a1 = d1 * inv * g1[l*2+1] + be1[l*2+1];
            // feed-forward with (fast, 1.5e-7-accurate) exact-form gelu
            float f0 = 0.0f, f1 = 0.0f;
#pragma unroll
            for (int j = 0; j < DFF; ++j) {
                float t = W1[l*64 + j*2+0]*a0 + W1[l*64 + j*2+1]*a1 + b1[l*32 + j];
                float g = gelu_exact(t);
                f0 += W2[l*64 +      j] * g;
                f1 += W2[l*64 + 32 + j] * g;
            }
            float z0 = a0 + f0 + b2[l*2+0];
            float z1 = a1 + f1 + b2[l*2+1];
            mu = 0.5f * (z0 + z1);
            d0 = z0 - mu; d1 = z1 - mu;
            inv = __builtin_amdgcn_rsqf(0.5f * (d0*d0 + d1*d1) + 1e-5f);
            h0[s] = d0 * inv * g2[l*2+0] + be2[l*2+0];
            h1[s] = d1 * inv * g2[l*2+1] + be2[l*2+1];
        }
    }

    // final = h[:, -1]
    sfinal[tid*2 + 0] = h0[SEQ-1];
    sfinal[tid*2 + 1] = h1[SEQ-1];
    __syncthreads();

    // ---------------- phase 2: vocab projection via V_WMMA_F32_16X16X4_F32 ---
    // D(16x16) = A(16x4, K=0..1 live) x B(4x16) + C(bias)
    const int wave    = tid >> 5;
    const int lane    = tid & 31;
    const int laneLow = lane & 15;
    const int hi      = lane >> 4;   // 0: lanes 0-15 (K=0,1 / rows 0-7), 1: lanes 16-31 (K=2,3 / rows 8-15)

#pragma unroll
    for (int rg = 0; rg < 2; ++rg) {
        const int rowBase = wave * 32 + rg * 16;      // row group within block
        const int r = rowBase + laneLow;
        // A fragment: lanes 0-15 carry K=0,1; lanes 16-31 (K=2,3) are zero padding
        const float fa0 = sfinal[r*2 + 0];
        const float fa1 = sfinal[r*2 + 1];
        v2f a;
        a.x = hi ? 0.0f : fa0;
        a.y = hi ? 0.0f : fa1;

        const int gRow = blockIdx.x * BLOCK + rowBase;
        // Throttled unroll: keep at most ~2 WMMA tiles (C+D fragments) live to
        // stay under 256 VGPRs and keep occupancy up for the HBM store stream.
#pragma unroll 2
        for (int nt = 0; nt < VOCAB / 16; ++nt) {
            const int col = nt * 16 + laneLow;
            // B fragment: lanes 0-15 hold rows K=0 (v0) and K=1 (v1); high lanes zero
            const float w0 = Wout[col*2 + 0];
            const float w1 = Wout[col*2 + 1];
            v2f bm;
            bm.x = hi ? 0.0f : w0;
            bm.y = hi ? 0.0f : w1;
            // C: bias broadcast down each column (same col index for both lane halves)
            const float bias = bout[col];
            v8f c;
#pragma unroll
            for (int i = 0; i < 8; ++i) c[i] = bias;

            v8f d = __builtin_amdgcn_wmma_f32_16x16x4_f32(
                /*neg_a=*/false, a, /*neg_b=*/false, bm,
                /*c_mod=*/(short)0, c, /*reuse_a=*/false, /*reuse_b=*/false);

            // D layout: VGPR i -> row i (lanes 0-15) / row i+8 (lanes 16-31), col = lane%16
#pragma unroll
            for (int i = 0; i < 8; ++i) {
                const int m = i + hi * 8;
                out[(size_t)(gRow + m) * VOCAB + nt * 16 + laneLow] = d[i];
            }
        }
    }
}

extern "C" void kernel_launch(void* const* d_in, const int* in_sizes, int n_in,
                              void* d_out, int out_size, void* d_ws, size_t ws_size,
                              hipStream_t stream) {
    const int*   x    = (const int*)  d_in[0];
    const float* Wqkv = (const float*)d_in[1];
    const float* bqkv = (const float*)d_in[2];
    const float* Wo   = (const float*)d_in[3];
    const float* bo   = (const float*)d_in[4];
    const float* W1   = (const float*)d_in[5];
    const float* b1   = (const float*)d_in[6];
    const float* W2   = (const float*)d_in[7];
    const float* b2   = (const float*)d_in[8];
    const float* g1   = (const float*)d_in[9];
    const float* be1  = (const float*)d_in[10];
    const float* g2   = (const float*)d_in[11];
    const float* be2  = (const float*)d_in[12];
    const float* Wout = (const float*)d_in[13];
    const float* bout = (const float*)d_in[14];
    float* out = (float*)d_out;

    const int batch = in_sizes[0] / SEQ;        // 131072
    dim3 grid(batch / BLOCK);                   // 512 blocks of 256 threads (8 waves)
    txf_fused_kernel<<<grid, BLOCK, 0, stream>>>(
        x, Wqkv, bqkv, Wo, bo, W1, b1, W2, b2, g1, be1, g2, be2, Wout, bout, out);
}